// AdapterPool_53180285059210
// MI455X (gfx1250) — compile-verified
//
#include <hip/hip_runtime.h>
#include <math.h>

// ---------------------------------------------------------------------------
// AdapterPool (L2P prompt selection) for MI455X / gfx1250
//
// Cost model: reading x_embed (402.7 MB) dominates -> bandwidth bound
// (~17us @ 23.3 TB/s). The similarity GEMM (256x100x768) is tiny, so we use
// the native f32 matrix unit (v_wmma_f32_16x16x4_f32) for exact-precision
// cosine similarities rather than downcasting.
//
// prompt_norm is padded to 112 rows (7 full 16-wide N tiles) with zero rows
// so the WMMA inner loop has NO predicated loads / EXEC churn.
// ---------------------------------------------------------------------------

typedef __attribute__((ext_vector_type(2))) float v2f;
typedef __attribute__((ext_vector_type(8))) float v8f;

#define BATCH   256
#define SEQ     512
#define DIM     768
#define POOL    100
#define POOLPAD 112   // 7 * 16, zero-padded rows 100..111
#define TOPK    5

// ---------------------------------------------------------------------------
// Kernel 1: per-batch max over sequence dim, fused with L2 normalization.
// One block per batch row; 192 threads x float4 = 768 floats per row.
// Streaming, fully coalesced global_load_b128; LDS tree-reduce for sum(x^2).
// ---------------------------------------------------------------------------
__global__ __launch_bounds__(192) void max_norm_kernel(
    const float* __restrict__ x, float* __restrict__ xn) {
  const int b = blockIdx.x;
  const int t = threadIdx.x;  // 0..191
  const float* base = x + (size_t)b * SEQ * DIM + t * 4;

  float4 m = make_float4(-INFINITY, -INFINITY, -INFINITY, -INFINITY);
  for (int s = 0; s < SEQ; ++s) {
    // speculative prefetch a few rows ahead (global_prefetch_b8)
    int sp = (s + 16 < SEQ) ? s + 16 : SEQ - 1;
    __builtin_prefetch(base + (size_t)sp * DIM, 0, 0);
    float4 v = *(const float4*)(base + (size_t)s * DIM);
    m.x = fmaxf(m.x, v.x);
    m.y = fmaxf(m.y, v.y);
    m.z = fmaxf(m.z, v.z);
    m.w = fmaxf(m.w, v.w);
  }

  __shared__ float red[192];
  red[t] = m.x * m.x + m.y * m.y + m.z * m.z + m.w * m.w;
  __syncthreads();
  for (int off = 96; off >= 3; off >>= 1) {   // 96,48,24,12,6,3
    if (t < off) red[t] += red[t + off];
    __syncthreads();
  }
  float total = red[0] + red[1] + red[2];
  float scale = rsqrtf(fmaxf(total, 1e-12f));

  float* o = xn + (size_t)b * DIM + t * 4;
  *(float4*)o = make_float4(m.x * scale, m.y * scale, m.z * scale, m.w * scale);
}

// ---------------------------------------------------------------------------
// Kernel 2: L2-normalize prompt_key rows into a 112-row padded buffer.
// Blocks >= POOL write zero rows (block-uniform branch, no divergence).
// ---------------------------------------------------------------------------
__global__ __launch_bounds__(192) void prompt_norm_kernel(
    const float* __restrict__ pk, float* __restrict__ pn) {
  const int p = blockIdx.x;   // 0..111
  const int t = threadIdx.x;
  float* o = pn + (size_t)p * DIM + t * 4;

  if (p >= POOL) {
    *(float4*)o = make_float4(0.0f, 0.0f, 0.0f, 0.0f);
    return;
  }

  float4 v = *(const float4*)(pk + (size_t)p * DIM + t * 4);

  __shared__ float red[192];
  red[t] = v.x * v.x + v.y * v.y + v.z * v.z + v.w * v.w;
  __syncthreads();
  for (int off = 96; off >= 3; off >>= 1) {
    if (t < off) red[t] += red[t + off];
    __syncthreads();
  }
  float total = red[0] + red[1] + red[2];
  float scale = rsqrtf(fmaxf(total, 1e-12f));

  *(float4*)o = make_float4(v.x * scale, v.y * scale, v.z * scale, v.w * scale);
}

// ---------------------------------------------------------------------------
// Kernel 3: similarity[256,100] = Xn @ Pn^T via V_WMMA_F32_16X16X4_F32.
// One wave (32 lanes) per 16x16 output tile; K=768 in 192 steps of 4.
// A(16x4): lane(0..15)=M row, lane-half selects K{0,1}/{2,3} -> contig float2.
// B(4x16): lane(0..15)=N col, same K striping -> contiguous float2 per row.
// D(16x16 f32): VGPR i = row (i + 8*half), col = lane&15.
// B rows 100..111 are physical zero rows -> branchless inner loop,
// EXEC stays all-1s throughout (WMMA requirement).
// ---------------------------------------------------------------------------
__global__ __launch_bounds__(32) void sim_wmma_kernel(
    const float* __restrict__ xn, const float* __restrict__ pn,
    float* __restrict__ sim) {
  const int mTile = blockIdx.x;       // 0..15
  const int nTile = blockIdx.y;       // 0..6
  const int lane  = threadIdx.x;      // 0..31
  const int hi    = lane >> 4;        // lane half
  const int lr    = lane & 15;
  const int mBase = mTile * 16;
  const int nBase = nTile * 16;
  const int prow  = nBase + lr;       // 0..111, always in-bounds of padded pn

  const float* arow = xn + (size_t)(mBase + lr) * DIM;
  const float* brow = pn + (size_t)prow * DIM;

  v8f c = {};
#pragma unroll 8
  for (int k0 = 0; k0 < DIM; k0 += 4) {
    const int kk = k0 + 2 * hi;
    float2 av = *(const float2*)(arow + kk);
    float2 bv = *(const float2*)(brow + kk);
    v2f a, b;
    a.x = av.x; a.y = av.y;
    b.x = bv.x; b.y = bv.y;
    // (neg_a, A, neg_b, B, c_mod, C, reuse_a, reuse_b)
    c = __builtin_amdgcn_wmma_f32_16x16x4_f32(
        false, a, false, b, (short)0, c, false, false);
  }

  if (prow < POOL) {
#pragma unroll
    for (int i = 0; i < 8; ++i) {
      const int m = mBase + i + 8 * hi;
      sim[(size_t)m * POOL + prow] = c[i];
    }
  }
}

// ---------------------------------------------------------------------------
// Kernel 4: stable top-5 per row (lower index wins ties, matching lax.top_k),
// emits int indices (ws), float indices (output), and per-row top-k sums.
// ---------------------------------------------------------------------------
__global__ __launch_bounds__(256) void topk_kernel(
    const float* __restrict__ sim, int* __restrict__ idxi,
    float* __restrict__ rowsum, float* __restrict__ idx_out) {
  const int row = blockIdx.x * blockDim.x + threadIdx.x;
  if (row >= BATCH) return;
  const float* s = sim + (size_t)row * POOL;

  int chosen[TOPK];
  float tsum = 0.0f;
  for (int k = 0; k < TOPK; ++k) {
    float best = -INFINITY;
    int bi = 0;
    for (int p = 0; p < POOL; ++p) {
      bool skip = false;
      for (int j = 0; j < k; ++j) skip = skip || (chosen[j] == p);
      float v = s[p];
      if (!skip && v > best) { best = v; bi = p; }
    }
    chosen[k] = bi;
    tsum += best;
    idxi[row * TOPK + k] = bi;
    idx_out[row * TOPK + k] = (float)bi;
  }
  rowsum[row] = tsum;
}

// ---------------------------------------------------------------------------
// Kernel 5: reduce_sim = sum(rowsum) / BATCH  (selected keys are unit-dotted
// against x_norm, so the sum equals the sum of the top-k similarities).
// ---------------------------------------------------------------------------
__global__ __launch_bounds__(256) void reduce_sim_kernel(
    const float* __restrict__ rowsum, float* __restrict__ out) {
  __shared__ float red[256];
  const int t = threadIdx.x;
  red[t] = rowsum[t];
  __syncthreads();
  for (int off = 128; off >= 1; off >>= 1) {
    if (t < off) red[t] += red[t + off];
    __syncthreads();
  }
  if (t == 0) out[0] = red[0] / (float)BATCH;
}

// ---------------------------------------------------------------------------
// Kernel 6: gather selected_key[b,k,:] = prompt_norm[idx[b,k],:]
// One block per (b,k); float4 copies.
// ---------------------------------------------------------------------------
__global__ __launch_bounds__(192) void gather_kernel(
    const float* __restrict__ pn, const int* __restrict__ idxi,
    float* __restrict__ sel) {
  const int bk = blockIdx.x;          // 0..1279
  const int t  = threadIdx.x;
  const int p  = idxi[bk];
  float4 v = *(const float4*)(pn + (size_t)p * DIM + t * 4);
  *(float4*)(sel + (size_t)bk * DIM + t * 4) = v;
}

// ---------------------------------------------------------------------------
// Launch
// d_out layout (floats): similarity[256*100] | selected_key[256*5*768] |
//                        reduce_sim[1] | idx[256*5]
// ---------------------------------------------------------------------------
extern "C" void kernel_launch(void* const* d_in, const int* in_sizes, int n_in,
                              void* d_out, int out_size, void* d_ws, size_t ws_size,
                              hipStream_t stream) {
  const float* x_embed    = (const float*)d_in[0];   // [256,512,768]
  const float* prompt_key = (const float*)d_in[1];   // [100,768]

  float* out       = (float*)d_out;
  float* sim_out   = out;                                      // 25600
  float* sel_out   = out + BATCH * POOL;                       // 983040
  float* rsum_out  = out + BATCH * POOL + BATCH * TOPK * DIM;  // 1
  float* idx_out   = rsum_out + 1;                             // 1280

  // workspace layout
  float* xn     = (float*)d_ws;                 // 256*768
  float* pn     = xn + BATCH * DIM;             // 112*768 (zero-padded tail)
  float* rowsum = pn + POOLPAD * DIM;           // 256
  int*   idxi   = (int*)(rowsum + BATCH);       // 256*5

  max_norm_kernel<<<BATCH, 192, 0, stream>>>(x_embed, xn);
  prompt_norm_kernel<<<POOLPAD, 192, 0, stream>>>(prompt_key, pn);
  sim_wmma_kernel<<<dim3(BATCH / 16, POOLPAD / 16), 32, 0, stream>>>(xn, pn, sim_out);
  topk_kernel<<<1, 256, 0, stream>>>(sim_out, idxi, rowsum, idx_out);
  reduce_sim_kernel<<<1, 256, 0, stream>>>(rowsum, rsum_out);
  gather_kernel<<<BATCH * TOPK, 192, 0, stream>>>(pn, idxi, sel_out);
}